// RNNTPrefixSearch_67310727463186
// MI455X (gfx1250) — compile-verified
//
#include <hip/hip_runtime.h>
#include <stdint.h>

typedef float v8f __attribute__((ext_vector_type(8)));
typedef float v2f __attribute__((ext_vector_type(2)));
typedef float f4v __attribute__((ext_vector_type(4)));

#define NEG_INF (-3.402823466e38f)
#define UDIM 32                      /* U fixed by the reference problem */

// ============================================================================
// Phase 1: per-(t,u) row logsumexp over D=4096, emit/blank gather.
// One 256-thread block per row; nontemporal b128 loads (stream-once data);
// wave reduction for the sum uses v_wmma_f32_16x16x4_f32 (A=ones, B=partials,
// C=0 -> D[m][n] = column sums of B; summing one D row over a 16-lane group
// gives the exact wave total, independent of B's lane striping).
// ============================================================================
__global__ __launch_bounds__(256) void rnnt_lse_kernel(
    const float* __restrict__ logits,
    const int*   __restrict__ targets,
    float* __restrict__ blk,
    float* __restrict__ emit,
    int D)
{
  __shared__ float red[16];
  const int row = blockIdx.x;
  const int tid = threadIdx.x;
  const int lane = tid & 31;
  const int wid  = tid >> 5;
  const size_t base = (size_t)row * (size_t)D;
  const f4v* p4 = (const f4v*)(logits + base);

  // 256 threads * 16 floats = 4096
  float vals[16];
  #pragma unroll
  for (int k = 0; k < 4; ++k) {
    f4v v = __builtin_nontemporal_load(&p4[k * 256 + tid]);
    vals[k * 4 + 0] = v.x;
    vals[k * 4 + 1] = v.y;
    vals[k * 4 + 2] = v.z;
    vals[k * 4 + 3] = v.w;
  }

  // ---- block max ----
  float tmax = NEG_INF;
  #pragma unroll
  for (int i = 0; i < 16; ++i) tmax = fmaxf(tmax, vals[i]);
  #pragma unroll
  for (int off = 16; off >= 1; off >>= 1)
    tmax = fmaxf(tmax, __shfl_xor(tmax, off, 32));
  if (lane == 0) red[wid] = tmax;
  __syncthreads();
  float bmax = NEG_INF;
  #pragma unroll
  for (int i = 0; i < 8; ++i) bmax = fmaxf(bmax, red[i]);

  // ---- block sum of exp(x - bmax) ----
  float tsum = 0.0f;
  #pragma unroll
  for (int i = 0; i < 16; ++i) tsum += __expf(vals[i] - bmax);

  // Wave-level sum via WMMA f32 16x16x4: D[m][n] = sum_k B[k][n]  (A=1, C=0).
  // Sum of all columns == sum of all B entries == wave total of tsum.
  v2f av; av.x = 1.0f; av.y = 1.0f;
  v2f bv; bv.x = tsum; bv.y = 0.0f;
  v8f cz = {0.f, 0.f, 0.f, 0.f, 0.f, 0.f, 0.f, 0.f};
  v8f dv = __builtin_amdgcn_wmma_f32_16x16x4_f32(
      false, av, false, bv, (short)0, cz, false, false);
  float w = dv[0];                 // lanes 0-15: D[0][n]; lanes 16-31: D[8][n] (equal)
  w += __shfl_xor(w, 1, 32);
  w += __shfl_xor(w, 2, 32);
  w += __shfl_xor(w, 4, 32);
  w += __shfl_xor(w, 8, 32);       // now: full wave total of tsum
  if (lane == 0) red[8 + wid] = w;
  __syncthreads();

  if (tid == 0) {
    float s = 0.0f;
    #pragma unroll
    for (int i = 0; i < 8; ++i) s += red[8 + i];
    float lse = bmax + __logf(s);
    int u = row & (UDIM - 1);          // U == 32 (power of two): no int division
    int tg = targets[u];
    float l0 = logits[base];           // BLANK = 0
    float lt = logits[base + tg];
    blk[row]  = l0 - lse;
    emit[row] = lt - lse;
  }
}

// ============================================================================
// Phase 2: anti-diagonal wavefront DP, 1 wave (32 lanes = 32 u values).
// blk/emit chunks (64 t-rows each, 16KB/chunk) are streamed into LDS with
// CDNA5 async global->LDS copies (ASYNCcnt), triple buffered so chunk c+2
// prefetch overlaps compute on chunks c / c+1.
// ============================================================================
#define CH 64
#define CHUNK_BYTES (CH * 32 * 4)     /* 8192 per array per chunk */

__device__ __forceinline__ int lidx(int t, int u) {
  int buf = (t >> 6) % 3;
  return buf * (CH * 32) + ((t & 63) << 5) + u;
}

__global__ __launch_bounds__(32) void rnnt_dp_kernel(
    const float* __restrict__ blkg,
    const float* __restrict__ emitg,
    const int*   __restrict__ tlens,
    float* __restrict__ out,
    int T, int NCH)
{
  __shared__ __align__(16) float sblk[3 * CH * 32];
  __shared__ __align__(16) float semt[3 * CH * 32];

  const int u  = threadIdx.x;
  const int U  = UDIM;
  const int tl = tlens[0];
  const int ND = T + U - 1;
  int* outi = (int*)out;

  const unsigned sblk_base = (unsigned)(uintptr_t)(&sblk[0]);
  const unsigned semt_base = (unsigned)(uintptr_t)(&semt[0]);
  const unsigned long long gb_blk  = (unsigned long long)(uintptr_t)blkg;
  const unsigned long long gb_emit = (unsigned long long)(uintptr_t)emitg;

  // 32 async b128 instructions per chunk (16 per array); lane moves 16B each.
  // Fully unrolled so the wave issues the whole chunk back-to-back (hardware
  // throttles at ASYNCcnt==63, which is fine).
  #define ISSUE_CHUNK(c)                                                      \
    do {                                                                      \
      unsigned buf   = (unsigned)((c) % 3);                                   \
      unsigned loffb = sblk_base + buf * CHUNK_BYTES + (unsigned)(u * 16);    \
      unsigned loffe = semt_base + buf * CHUNK_BYTES + (unsigned)(u * 16);    \
      int goff = (c) * CHUNK_BYTES + u * 16;                                  \
      _Pragma("unroll")                                                       \
      for (int j = 0; j < 16; ++j) {                                          \
        asm volatile("global_load_async_to_lds_b128 %0, %1, %2"               \
                     :: "v"(loffb), "v"(goff), "s"(gb_blk) : "memory");       \
        asm volatile("global_load_async_to_lds_b128 %0, %1, %2"               \
                     :: "v"(loffe), "v"(goff), "s"(gb_emit) : "memory");      \
        loffb += 512; loffe += 512; goff += 512;                              \
      }                                                                       \
    } while (0)

  int next = 0;
  int maxpre = NCH < 3 ? NCH : 3;
  for (; next < maxpre; ++next) ISSUE_CHUNK(next);

  float    a_self = NEG_INF;
  unsigned p_self = 0;           // pack: start<<16 | total

  for (int c = 0; c < NCH; ++c) {
    // chunks <= c+1 must be complete; at most one chunk (32 ops) may remain in flight
    if (next - (c + 2) >= 1) {
      asm volatile("s_wait_asynccnt 32" ::: "memory");
    } else {
      asm volatile("s_wait_asynccnt 0" ::: "memory");
    }

    int d_lo = (c == 0) ? 0 : c * CH + 32;
    int d_hi = (c + 1) * CH + 32;
    if (d_hi > ND) d_hi = ND;

    for (int d = d_lo; d < d_hi; ++d) {
      float    a_left = __shfl_up(a_self, 1, 32);
      unsigned p_left = __shfl_up(p_self, 1, 32);
      int t = d - u;

      float a_new; unsigned p_new;
      if (u == 0) {
        a_new = 0.0f;                                  // alpha[t,0] = 0
        p_new = ((unsigned)t << 16) | 1u;              // start=t, total=1
      } else {
        int tm = (t < 1) ? 0 : (t - 1);
        int tc = (t < 0) ? 0 : t;
        float bvv  = sblk[lidx(tm, u)];                // blk[t-1, u]
        float evv  = semt[lidx(tc, u - 1)];            // emit[t, u-1]
        float stay = (t >= 1) ? (a_self + bvv) : NEG_INF;
        float emv  = a_left + evv;
        bool  takeStay = (stay >= emv);                // tie prefers blank path
        a_new = takeStay ? stay : emv;
        p_new = (takeStay ? p_self : p_left) + 1u;     // total += 1
      }

      if (t >= 0 && t < T) {
        a_self = a_new;
        p_self = p_new;
        if (u == tl) {
          float la = a_new + sblk[lidx(t, u)];         // + blk[t, tl]
          out[1 + t]           = la;
          outi[1 + T + t]      = (int)(p_new >> 16);           // start
          outi[1 + 2 * T + t]  = (int)(p_new & 0xffffu) + 1;   // total + 1
          if (t == T - 1) out[0] = la;                 // la[-1]
        }
      }
    }

    if (next < NCH) { ISSUE_CHUNK(next); ++next; }
  }
  #undef ISSUE_CHUNK
}

// ============================================================================
extern "C" void kernel_launch(void* const* d_in, const int* in_sizes, int n_in,
                              void* d_out, int out_size, void* d_ws, size_t ws_size,
                              hipStream_t stream) {
  const float* logits  = (const float*)d_in[0];
  const int*   targets = (const int*)d_in[1];
  const int*   tlens   = (const int*)d_in[3];

  const int U = UDIM;                        // 32 (== in_sizes[1])
  const int D = 4096;                        // vocab (fixed by reference)
  const long long tot = (long long)in_sizes[0];
  const int T = (int)(tot / ((long long)U * (long long)D));   // 2000

  const int NCH = (T + CH - 1) / CH;         // 32 chunks of 64 rows
  const int TP  = NCH * CH;                  // padded rows (2048)
  (void)TP; (void)ws_size; (void)n_in; (void)out_size;

  float* blk  = (float*)d_ws;                // TP*U floats
  float* emit = blk + (size_t)TP * U;        // TP*U floats  (total 512KB)

  rnnt_lse_kernel<<<T * U, 256, 0, stream>>>(logits, targets, blk, emit, D);
  rnnt_dp_kernel<<<1, 32, 0, stream>>>(blk, emit, tlens, (float*)d_out, T, NCH);
}